// GCN_CORA_30374008717765
// MI455X (gfx1250) — compile-verified
//
#include <hip/hip_runtime.h>
#include <math.h>

#define N_NODES 100000
#define D_IN    1433
#define HID     80
#define N_CLS   7

typedef __attribute__((ext_vector_type(2))) float v2f;
typedef __attribute__((ext_vector_type(8))) float v8f;

// ---------------------------------------------------------------- utilities
__global__ void fill4_kernel(float4* __restrict__ p, float v, int n4) {
    int i = blockIdx.x * blockDim.x + threadIdx.x;
    if (i < n4) p[i] = make_float4(v, v, v, v);
}

// deg[dst] += 1 for every edge (deg pre-filled with 1.0 for the self loop)
__global__ void deg_kernel(const int* __restrict__ dst, float* __restrict__ deg, int E) {
    int e = blockIdx.x * blockDim.x + threadIdx.x;
    if (e < E) atomicAdd(&deg[dst[e]], 1.0f);
}

__global__ void dinv_kernel(const float* __restrict__ deg, float* __restrict__ dinv, int n) {
    int i = blockIdx.x * blockDim.x + threadIdx.x;
    if (i < n) {
        float d = deg[i];
        dinv[i] = (d > 0.0f) ? rsqrtf(d) : 0.0f;
    }
}

// ------------------------------------------------- WMMA fp32 GEMM  C = A*B
// A: M x K (row major), B: K x 80 (row major), C: M x 80.
// Block = 256 threads = 8 waves; each wave computes a 16x80 strip of C using
// five 16x16 accumulators and V_WMMA_F32_16X16X4_F32.  B is staged through
// LDS in 32x80 panels (b128 loads/stores, shared by all 8 waves).  Full
// K-chunks run guard-free; only the final partial chunk carries selects.
template <int K>
__global__ __launch_bounds__(256) void gemm_wmma_kernel(
    const float* __restrict__ A, const float* __restrict__ B,
    float* __restrict__ C, int M) {
    __shared__ float Bs[32 * HID];   // 10 KB

    const int tid  = threadIdx.x;
    const int lane = tid & 31;
    const int wid  = tid >> 5;
    const int col  = lane & 15;      // N index within a 16-wide tile / M row select
    const int hk   = lane >> 4;      // half-wave selector
    const int m0   = blockIdx.x * 128 + wid * 16;
    const bool active = (m0 + 16) <= M;

    // Clamped in-bounds pointer: inactive waves compute on garbage that is
    // never stored, so no per-load predication is needed in the fast path.
    const float* arow = A + (size_t)(active ? (m0 + col) : 0) * K;

    v8f acc[5] = {};

    constexpr int KFULL = (K / 32) * 32;

    for (int kc = 0; kc < KFULL; kc += 32) {
        // Guard-free b128 staging: 32*80 floats = 640 float4s by 256 threads.
        const float4* Bg  = (const float4*)(B + (size_t)kc * HID);
        float4*       Bs4 = (float4*)Bs;
        Bs4[tid]       = Bg[tid];
        Bs4[tid + 256] = Bg[tid + 256];
        if (tid < 128) Bs4[tid + 512] = Bg[tid + 512];
        __syncthreads();

        if (kc + 32 < K) __builtin_prefetch(arow + kc + 32, 0, 0);

#pragma unroll
        for (int kk = 0; kk < 32; kk += 4) {
            // A fragment (16x4 f32): lanes 0-15 hold K=k,k+1; lanes 16-31 K=k+2,k+3
            int kA = kc + kk + 2 * hk;
            v2f a;
            a.x = arow[kA];
            a.y = arow[kA + 1];

            int kb = kk + 2 * hk;   // B fragment rows mirror the A striping
#pragma unroll
            for (int t = 0; t < 5; ++t) {
                v2f b;
                b.x = Bs[ kb      * HID + t * 16 + col];
                b.y = Bs[(kb + 1) * HID + t * 16 + col];
                acc[t] = __builtin_amdgcn_wmma_f32_16x16x4_f32(
                    false, a, false, b, (short)0, acc[t], false, false);
            }
        }
        __syncthreads();
    }

    if constexpr (KFULL < K) {
        // Final partial chunk, zero-padded in LDS and guarded A loads.
        for (int i = tid; i < 32 * HID; i += 256) {
            int kk = i / HID;
            int n  = i - kk * HID;
            int k  = KFULL + kk;
            Bs[i] = (k < K) ? B[(size_t)k * HID + n] : 0.0f;
        }
        __syncthreads();

#pragma unroll
        for (int kk = 0; kk < 32; kk += 4) {
            int kA = KFULL + kk + 2 * hk;
            v2f a;
            a.x = (kA     < K) ? arow[kA]     : 0.0f;
            a.y = (kA + 1 < K) ? arow[kA + 1] : 0.0f;

            int kb = kk + 2 * hk;
#pragma unroll
            for (int t = 0; t < 5; ++t) {
                v2f b;
                b.x = Bs[ kb      * HID + t * 16 + col];
                b.y = Bs[(kb + 1) * HID + t * 16 + col];
                acc[t] = __builtin_amdgcn_wmma_f32_16x16x4_f32(
                    false, a, false, b, (short)0, acc[t], false, false);
            }
        }
    }

    if (active) {
        // C/D layout: VGPR r holds rows r (lanes 0-15) and r+8 (lanes 16-31)
#pragma unroll
        for (int t = 0; t < 5; ++t) {
#pragma unroll
            for (int r = 0; r < 8; ++r) {
                C[(size_t)(m0 + r + 8 * hk) * HID + t * 16 + col] = acc[t][r];
            }
        }
    }
}

// ----------------------------------------- edge aggregation (segment_sum)
// One thread per (edge, 4-feature chunk): b128 gather of the L2-resident
// T[src] row, four atomic f32 adds into contiguous addresses of A[dst].
__global__ void scatter_kernel(const float* __restrict__ T,
                               const int* __restrict__ src,
                               const int* __restrict__ dst,
                               const float* __restrict__ dinv,
                               float* __restrict__ Acc, int E) {
    const int C4 = HID / 4;          // 20 float4 chunks per row
    int i = blockIdx.x * blockDim.x + threadIdx.x;
    int total = E * C4;
    if (i >= total) return;
    int e  = i / C4;
    int f4 = i - e * C4;
    int s = src[e];
    int d = dst[e];
    float w = dinv[s] * dinv[d];
    float4 v = ((const float4*)(T + (size_t)s * HID))[f4];
    float* ap = Acc + (size_t)d * HID + f4 * 4;
    atomicAdd(ap + 0, v.x * w);
    atomicAdd(ap + 1, v.y * w);
    atomicAdd(ap + 2, v.z * w);
    atomicAdd(ap + 3, v.w * w);
}

// Self-loop contribution (norm = dinv[i]^2) + bias + ReLU, in place in Acc.
__global__ void bias_relu_kernel(float* __restrict__ Acc,
                                 const float* __restrict__ T,
                                 const float* __restrict__ dinv,
                                 const float* __restrict__ bias, int n4) {
    const int C4 = HID / 4;
    int i = blockIdx.x * blockDim.x + threadIdx.x;
    if (i >= n4) return;
    int node = i / C4;
    int f4   = i - node * C4;
    float di = dinv[node];
    float w  = di * di;
    float4 t = ((const float4*)T)[i];
    float4 a = ((float4*)Acc)[i];
    float4 b = ((const float4*)bias)[f4];
    float4 r;
    r.x = fmaxf(a.x + t.x * w + b.x, 0.0f);
    r.y = fmaxf(a.y + t.y * w + b.y, 0.0f);
    r.z = fmaxf(a.z + t.z * w + b.z, 0.0f);
    r.w = fmaxf(a.w + t.w * w + b.w, 0.0f);
    ((float4*)Acc)[i] = r;
}

// ---------------------------------- head: logits = H @ Wl + bl, log_softmax
__global__ void head_kernel(const float* __restrict__ H,
                            const float* __restrict__ Wl,
                            const float* __restrict__ bl,
                            float* __restrict__ out, int n) {
    int i = blockIdx.x * blockDim.x + threadIdx.x;
    if (i >= n) return;
    const float* h = H + (size_t)i * HID;
    float acc[N_CLS];
#pragma unroll
    for (int c = 0; c < N_CLS; ++c) acc[c] = bl[c];
    for (int f = 0; f < HID; ++f) {
        float hv = h[f];
#pragma unroll
        for (int c = 0; c < N_CLS; ++c) acc[c] += hv * Wl[f * N_CLS + c];
    }
    float m = acc[0];
#pragma unroll
    for (int c = 1; c < N_CLS; ++c) m = fmaxf(m, acc[c]);
    float s = 0.0f;
#pragma unroll
    for (int c = 0; c < N_CLS; ++c) s += expf(acc[c] - m);
    float lse = m + logf(s);
#pragma unroll
    for (int c = 0; c < N_CLS; ++c) out[(size_t)i * N_CLS + c] = acc[c] - lse;
}

// -------------------------------------------------------------------- driver
extern "C" void kernel_launch(void* const* d_in, const int* in_sizes, int n_in,
                              void* d_out, int out_size, void* d_ws, size_t ws_size,
                              hipStream_t stream) {
    const float* x   = (const float*)d_in[0];
    const int*   ei  = (const int*)d_in[1];   // (2, E) int32 (JAX x64 disabled)
    const float* W1  = (const float*)d_in[2];
    const float* b1  = (const float*)d_in[3];
    const float* W2  = (const float*)d_in[4];
    const float* b2  = (const float*)d_in[5];
    const float* Wl  = (const float*)d_in[6];
    const float* bl  = (const float*)d_in[7];
    float* out = (float*)d_out;

    const int N = N_NODES;
    const int E = in_sizes[1] / 2;
    const int* src = ei;
    const int* dst = ei + E;

    // Workspace layout (floats): deg[N] | dinv[N] | T[N*80] | Acc[N*80]
    float* ws   = (float*)d_ws;
    float* deg  = ws;
    float* dinv = ws + N;
    float* T    = ws + 2 * (size_t)N;
    float* Acc  = T + (size_t)N * HID;

    const int TB = 256;
    const int C4 = HID / 4;
    int nb_nodes = (N + TB - 1) / TB;
    int nb_deg4  = (N / 4 + TB - 1) / TB;
    int nb_acc4  = (N * C4 + TB - 1) / TB;
    int nb_edge  = (E + TB - 1) / TB;
    int nb_scat  = (E * C4 + TB - 1) / TB;
    int nb_gemm  = (N + 127) / 128;

    // --- normalization: deg (with self loop), dinv = rsqrt(deg)
    fill4_kernel<<<nb_deg4, TB, 0, stream>>>((float4*)deg, 1.0f, N / 4);
    deg_kernel<<<nb_edge, TB, 0, stream>>>(dst, deg, E);
    dinv_kernel<<<nb_nodes, TB, 0, stream>>>(deg, dinv, N);

    // --- layer 1: T = x @ W1 ; Acc = segsum(T[src]*norm) ; H1 = relu(Acc + self + b1)
    gemm_wmma_kernel<D_IN><<<nb_gemm, TB, 0, stream>>>(x, W1, T, N);
    fill4_kernel<<<nb_acc4, TB, 0, stream>>>((float4*)Acc, 0.0f, N * C4);
    scatter_kernel<<<nb_scat, TB, 0, stream>>>(T, src, dst, dinv, Acc, E);
    bias_relu_kernel<<<nb_acc4, TB, 0, stream>>>(Acc, T, dinv, b1, N * C4);

    // --- layer 2: T = H1 @ W2 ; Acc = segsum ; H2 = relu(Acc + self + b2)
    gemm_wmma_kernel<HID><<<nb_gemm, TB, 0, stream>>>(Acc, W2, T, N);
    fill4_kernel<<<nb_acc4, TB, 0, stream>>>((float4*)Acc, 0.0f, N * C4);
    scatter_kernel<<<nb_scat, TB, 0, stream>>>(T, src, dst, dinv, Acc, E);
    bias_relu_kernel<<<nb_acc4, TB, 0, stream>>>(Acc, T, dinv, b2, N * C4);

    // --- head: log_softmax(H2 @ Wl + bl)
    head_kernel<<<nb_nodes, TB, 0, stream>>>(Acc, Wl, bl, out, N);
}